// HypAgg_pyg_89996744721065
// MI455X (gfx1250) — compile-verified
//
#include <hip/hip_runtime.h>
#include <hip/hip_bf16.h>
#include <math.h>

typedef __attribute__((ext_vector_type(2))) float v2f;
typedef __attribute__((ext_vector_type(8))) float v8f;

#define FEAT 64
#define MIN_NORM 1e-15f
#define PROJ_EPS 4e-3f

// ---------------------------------------------------------------------------
// 1. init: zero the scatter accumulator (d_out) and set deg = 1 (self loops)
// ---------------------------------------------------------------------------
__global__ void init_kernel(float* __restrict__ out, float* __restrict__ deg,
                            int n_nodes, int total) {
  int i = blockIdx.x * blockDim.x + threadIdx.x;
  if (i < total)   out[i] = 0.0f;
  if (i < n_nodes) deg[i] = 1.0f;   // self-loop contributes 1 to every node
}

// ---------------------------------------------------------------------------
// 2. degree count over destinations (atomics resolve in L2 — deg is 400 KB)
// ---------------------------------------------------------------------------
__global__ void degree_kernel(const int* __restrict__ ei, float* __restrict__ deg,
                              int ne) {
  int e = blockIdx.x * blockDim.x + threadIdx.x;
  if (e < ne) atomicAdd(&deg[ei[ne + e]], 1.0f);   // ei layout: [2, E]; dst row
}

// ---------------------------------------------------------------------------
// 3. deg -> deg^-1/2 in place (deg >= 1 always, so no zero guard needed)
// ---------------------------------------------------------------------------
__global__ void dinv_kernel(float* __restrict__ deg, int n_nodes) {
  int i = blockIdx.x * blockDim.x + threadIdx.x;
  if (i < n_nodes) deg[i] = rsqrtf(deg[i]);
}

// ---------------------------------------------------------------------------
// 4. h = x @ W  with V_WMMA_F32_16X16X4_F32 (exact fp32 GEMM on the XDL path)
//    One wave32 per 16x16 output tile; K=64 -> 16 WMMA steps of K=4.
//    A 16x4 f32 layout (ISA 7.12.2): lanes 0-15 hold M=lane, {K0,K1} in v0,v1;
//    lanes 16-31 hold M=lane-16, {K2,K3}. B 4x16 analogous (rows split by lane
//    half). C/D: VGPR r = row M=r (lanes 0-15) / M=r+8 (lanes 16-31).
// ---------------------------------------------------------------------------
__global__ void gemm_wmma_f32(const float* __restrict__ x, const float* __restrict__ W,
                              float* __restrict__ h, int rowblks) {
  int wid  = (blockIdx.x * blockDim.x + threadIdx.x) >> 5;
  int lane = threadIdx.x & 31;
  int rowblk = wid >> 2;            // which 16-row block of x
  int nt     = wid & 3;             // which 16-col tile of W (FEAT/16 = 4)
  if (rowblk >= rowblks) return;    // wave-uniform: EXEC stays all-ones

  int m0   = rowblk * 16;
  int n0   = nt * 16;
  int half = lane >> 4;             // 0: lanes 0-15, 1: lanes 16-31
  int l16  = lane & 15;

  v8f c = {};
#pragma unroll
  for (int kb = 0; kb < 16; ++kb) {
    int k0 = kb * 4;
    v2f a, b;
    const float* xr = x + (m0 + l16) * FEAT + k0 + 2 * half;
    a.x = xr[0];                                   // K = k0 + 2*half
    a.y = xr[1];                                   // K = k0 + 2*half + 1
    b.x = W[(k0 + 2 * half)     * FEAT + n0 + l16];
    b.y = W[(k0 + 2 * half + 1) * FEAT + n0 + l16];
    // 8 args: (neg_a, A, neg_b, B, c_mod, C, reuse_a, reuse_b)
    c = __builtin_amdgcn_wmma_f32_16x16x4_f32(false, a, false, b,
                                              (short)0, c, false, false);
  }
#pragma unroll
  for (int r = 0; r < 8; ++r)
    h[(m0 + r + 8 * half) * FEAT + n0 + l16] = c[r];
}

// ---------------------------------------------------------------------------
// 5. edge scatter: out[dst] += h[src] * dinv[src]*dinv[dst]
//    64 threads per edge (one per feature). h and out both fit in the 192 MB
//    L2, so the random gather + atomic scatter stay on-chip.
// ---------------------------------------------------------------------------
__global__ void scatter_edges(const float* __restrict__ h,
                              const float* __restrict__ dinv,
                              const int* __restrict__ ei,
                              float* __restrict__ out, int ne) {
  int t = blockIdx.x * blockDim.x + threadIdx.x;
  int e = t >> 6;
  int j = t & 63;
  if (e >= ne) return;
  int s = ei[e];          // src row of edge_index
  int d = ei[ne + e];     // dst row
  float w = dinv[s] * dinv[d];
  atomicAdd(&out[d * FEAT + j], h[s * FEAT + j] * w);
}

// ---------------------------------------------------------------------------
// 6. finalize: u = out[n] + h[n]*dinv[n]^2 + b ; expmap0 (c=1) ; proj
//    One wave32 per node, each lane owns 2 features; norm via shfl_xor
//    butterfly (wave32-native cross-lane reduce). In-place on d_out.
// ---------------------------------------------------------------------------
__global__ void finalize_kernel(const float* __restrict__ h,
                                const float* __restrict__ dinv,
                                const float* __restrict__ bias,
                                float* __restrict__ out, int n_nodes) {
  int node = (blockIdx.x * blockDim.x + threadIdx.x) >> 5;
  int lane = threadIdx.x & 31;
  if (node >= n_nodes) return;

  float w2 = dinv[node];
  w2 = w2 * w2;                      // self-loop norm: dinv[n]*dinv[n]
  int base = node * FEAT + 2 * lane;
  float ux = out[base]     + h[base]     * w2 + bias[2 * lane];
  float uy = out[base + 1] + h[base + 1] * w2 + bias[2 * lane + 1];

  float ss = ux * ux + uy * uy;
#pragma unroll
  for (int off = 16; off > 0; off >>= 1)
    ss += __shfl_xor(ss, off, 32);

  float un = fmaxf(sqrtf(ss), MIN_NORM);
  float th = tanhf(un);              // sqrt(c) = 1
  float scale = th / un;             // expmap0: tanh(||u||) * u / ||u||

  // proj: resulting norm is tanh(||u||); clip to (1 - PROJ_EPS)
  float vn      = fmaxf(th, MIN_NORM);
  float maxnorm = 1.0f - PROJ_EPS;
  float pscale  = (vn > maxnorm) ? (maxnorm / vn) : 1.0f;
  scale *= pscale;

  out[base]     = ux * scale;
  out[base + 1] = uy * scale;
}

// ---------------------------------------------------------------------------
extern "C" void kernel_launch(void* const* d_in, const int* in_sizes, int n_in,
                              void* d_out, int out_size, void* d_ws, size_t ws_size,
                              hipStream_t stream) {
  const float* x  = (const float*)d_in[0];   // [N, 64]
  const float* W  = (const float*)d_in[1];   // [64, 64]
  const float* b  = (const float*)d_in[2];   // [64]
  const int*   ei = (const int*)d_in[3];     // [2, E]

  const int n_nodes = in_sizes[0] / FEAT;    // 100000
  const int ne      = in_sizes[3] / 2;       // 1200000
  float* out = (float*)d_out;                // [N, 64]

  // workspace layout: h [N*64 floats] | deg/dinv [N floats]
  float* h   = (float*)d_ws;
  float* deg = h + (size_t)n_nodes * FEAT;

  const int total = n_nodes * FEAT;

  // 1. zero accumulator, deg = 1
  init_kernel<<<(total + 255) / 256, 256, 0, stream>>>(out, deg, n_nodes, total);

  // 2. degree over destinations
  degree_kernel<<<(ne + 255) / 256, 256, 0, stream>>>(ei, deg, ne);

  // 3. deg -> deg^-1/2
  dinv_kernel<<<(n_nodes + 255) / 256, 256, 0, stream>>>(deg, n_nodes);

  // 4. h = x @ W via fp32 WMMA (N = 100000 is an exact multiple of 16)
  {
    int rowblks = (n_nodes + 15) / 16;
    int waves   = rowblks * (FEAT / 16);
    int threads = waves * 32;
    gemm_wmma_f32<<<(threads + 255) / 256, 256, 0, stream>>>(x, W, h, rowblks);
  }

  // 5. edge scatter-add (L2-resident atomics)
  {
    long long threads = (long long)ne * FEAT;
    int blocks = (int)((threads + 255) / 256);
    scatter_edges<<<blocks, 256, 0, stream>>>(h, deg, ei, out, ne);
  }

  // 6. self-loop + bias + expmap0 + proj, one wave per node
  {
    long long threads = (long long)n_nodes * 32;
    int blocks = (int)((threads + 255) / 256);
    finalize_kernel<<<blocks, 256, 0, stream>>>(h, deg, b, out, n_nodes);
  }
}